// RNN_52475910423250
// MI455X (gfx1250) — compile-verified
//
#include <hip/hip_runtime.h>
#include <hip/hip_bf16.h>
#include <math.h>

#define T_DIM 32768
#define I_DIM 256
#define H_DIM 1024
#define O_DIM 256

#define NBLK 16                       // workgroups cooperating on the scan
#define ROWS_PER_BLK (H_DIM / NBLK)   // 64 rows of Wh per WGP, resident in LDS
#define SCAN_THREADS 256

typedef __attribute__((ext_vector_type(2))) float v2f;
typedef __attribute__((ext_vector_type(4))) float v4f;
typedef __attribute__((ext_vector_type(8))) float v8f;
typedef int v4i __attribute__((vector_size(16)));

// ---- CDNA5 async Global->LDS copy (ASYNCcnt path), guarded ----------------
#if __has_builtin(__builtin_amdgcn_global_load_async_to_lds_b128) && \
    __has_builtin(__builtin_amdgcn_s_wait_asynccnt)
#define HAVE_ASYNC_LDS 1
typedef __attribute__((address_space(1))) v4i as1_v4i;   // global int32x4
typedef __attribute__((address_space(3))) v4i as3_v4i;   // LDS int32x4
__device__ __forceinline__ void async_copy_b128(const void* gsrc, void* ldst) {
  __builtin_amdgcn_global_load_async_to_lds_b128((as1_v4i*)gsrc,
                                                 (as3_v4i*)ldst, 0, 0);
}
__device__ __forceinline__ void async_wait0() {
  __builtin_amdgcn_s_wait_asynccnt(0);
}
#endif

// ---------------------------------------------------------------------------
// Phase 1: xw[t][h] = sum_i x[t][i] * Wx[h][i] + bh[h]     (T x H output)
// One wave32 per 16x16 tile, V_WMMA_F32_16X16X4_F32, K looped 256/4 = 64x.
// ---------------------------------------------------------------------------
#if __has_builtin(__builtin_amdgcn_wmma_f32_16x16x4_f32)

__global__ void __launch_bounds__(256)
xw_gemm(const float* __restrict__ x, const float* __restrict__ Wx,
        const float* __restrict__ bh, float* __restrict__ xw) {
  const int wave = threadIdx.x >> 5;
  const int lane = threadIdx.x & 31;
  const int tile = blockIdx.x * 8 + wave;        // (T/16)*(H/16) tiles, t-major
  const int tiles_h = H_DIM / 16;                // 64
  const int t0 = (tile / tiles_h) * 16;
  const int h0 = (tile % tiles_h) * 16;
  const int l15 = lane & 15;
  const int hi  = lane >> 4;

  const float* xrow = x  + (size_t)(t0 + l15) * I_DIM + 2 * hi;  // A frag base
  const float* wrow = Wx + (size_t)(h0 + l15) * I_DIM + 2 * hi;  // B frag base

  v8f acc = {};
#pragma unroll 4
  for (int k = 0; k < I_DIM; k += 4) {
    v2f a = *(const v2f*)(xrow + k);
    v2f b = *(const v2f*)(wrow + k);
    acc = __builtin_amdgcn_wmma_f32_16x16x4_f32(
        /*neg_a=*/false, a, /*neg_b=*/false, b,
        /*c_mod=*/(short)0, acc, /*reuse_a=*/false, /*reuse_b=*/false);
  }

  const float bias = bh[h0 + l15];
  float* dst = xw + (size_t)(t0 + 8 * hi) * H_DIM + h0 + l15;
#pragma unroll
  for (int r = 0; r < 8; ++r) dst[(size_t)r * H_DIM] = acc[r] + bias;
}
#define XW_GRID ((T_DIM / 16) * (H_DIM / 16) / 8)

#else  // fallback: plain VALU (only if the f32 WMMA builtin is absent)

__global__ void __launch_bounds__(256)
xw_gemm(const float* __restrict__ x, const float* __restrict__ Wx,
        const float* __restrict__ bh, float* __restrict__ xw) {
  const size_t idx = (size_t)blockIdx.x * 256 + threadIdx.x;
  const int t = (int)(idx / H_DIM), h = (int)(idx % H_DIM);
  float s = bh[h];
  const float* xr = x + (size_t)t * I_DIM;
  const float* wr = Wx + (size_t)h * I_DIM;
#pragma unroll 8
  for (int k = 0; k < I_DIM; ++k) s = fmaf(xr[k], wr[k], s);
  xw[idx] = s;
}
#define XW_GRID ((T_DIM * H_DIM) / 256)
#endif

// ---------------------------------------------------------------------------
// Phase 2: sequential scan.  16 WGPs; block b holds Wh rows [64b,64b+64) in
// 256KB dynamic LDS (of 320KB).  Each step: 64x1024 matvec from LDS, tanh,
// publish 64-slice of h via L2, grid-barrier, async-gather full h into LDS.
// ---------------------------------------------------------------------------
__device__ __forceinline__ void grid_sync(int* __restrict__ bar, int step,
                                          bool in_cluster) {
  __threadfence();            // release: every wave flushes its stores
  __syncthreads();
  if (in_cluster) {
    // CDNA5 HW cluster barrier (-3); NOP if not cluster-dispatched.
    __builtin_amdgcn_s_cluster_barrier();
  }
  if (threadIdx.x == 0) {
    const int arrive = atomicAdd(&bar[0], 1);          // monotonic counter
    if (arrive == NBLK * (step + 1) - 1) {
      atomicExch(&bar[1], step + 1);                   // release generation
    } else {
      while (atomicAdd(&bar[1], 0) < step + 1) __builtin_amdgcn_s_sleep(1);
    }
  }
  __syncthreads();
  __threadfence();            // acquire: every wave invalidates before re-read
}

__global__ void __launch_bounds__(SCAN_THREADS, 1)
rnn_scan(const float* __restrict__ xw, const float* __restrict__ Wh,
         float* __restrict__ hs_out,            // [H, T]
         float* __restrict__ hshare,            // [2][H] in L2
         int* __restrict__ bar) {
  extern __shared__ float lds[];
  float* whs  = lds;                            // 64*1024 f32 = 256KB
  float* hbuf = lds + ROWS_PER_BLK * H_DIM;     // 1024 f32

  const int b = blockIdx.x, tid = threadIdx.x;

  // Stage this WGP's Wh slice into LDS (resident for all 32768 steps).
  const size_t wbase = (size_t)b * ROWS_PER_BLK * H_DIM;
#ifdef HAVE_ASYNC_LDS
  for (int i = tid * 4; i < ROWS_PER_BLK * H_DIM; i += SCAN_THREADS * 4)
    async_copy_b128(&Wh[wbase + i], &whs[i]);   // 16B per lane per issue
#else
  for (int i = tid; i < ROWS_PER_BLK * H_DIM; i += SCAN_THREADS)
    whs[i] = Wh[wbase + i];
#endif
  for (int i = tid; i < H_DIM; i += SCAN_THREADS) hbuf[i] = 0.0f;  // h0 = 0
#ifdef HAVE_ASYNC_LDS
  async_wait0();
#endif
  __syncthreads();

  const int row = tid >> 2;                     // 0..63  (4 threads per row)
  const int seg = tid & 3;                      // each sums 256 of 1024 cols
  const int h   = b * ROWS_PER_BLK + row;
  const v4f* w4 = (const v4f*)(whs + row * H_DIM + (seg << 8));
  const v4f* h4 = (const v4f*)(hbuf + (seg << 8));
  float* out_row = hs_out + (size_t)h * T_DIM;

  const bool in_cluster = (__builtin_amdgcn_cluster_id_x() != 0);

  for (int t = 0; t < T_DIM; ++t) {
    if (seg == 0 && t + 4 < T_DIM)              // warm next xw slice into L2
      __builtin_prefetch(&xw[(size_t)(t + 4) * H_DIM + h], 0, 0);

    float s = 0.0f;
#pragma unroll 16
    for (int k = 0; k < 64; ++k) {              // 64 x ds_load_b128 per stream
      v4f w = w4[k], hv = h4[k];
      s = fmaf(w.x, hv.x, s); s = fmaf(w.y, hv.y, s);
      s = fmaf(w.z, hv.z, s); s = fmaf(w.w, hv.w, s);
    }
    // reduce the 4 partial sums (lanes 4r..4r+3 of the same wave32)
    s += __shfl_xor(s, 1, 32);
    s += __shfl_xor(s, 2, 32);

    if (seg == 0) {
      const float hn = tanhf(xw[(size_t)t * H_DIM + h] + s);
      out_row[t] = hn;                          // hidden_states[h, t]
      hshare[(t & 1) * H_DIM + h] = hn;         // publish slice (double buf)
    }
    grid_sync(bar, t, in_cluster);

    // gather the full new h into LDS for the next step (4KB per WGP)
    const float* src = hshare + (t & 1) * H_DIM;
#ifdef HAVE_ASYNC_LDS
    async_copy_b128(&src[tid * 4], &hbuf[tid * 4]);   // 256 lanes x 16B = 4KB
    async_wait0();
#else
    for (int i = tid; i < H_DIM; i += SCAN_THREADS) hbuf[i] = src[i];
#endif
    __syncthreads();
  }
}

// ---------------------------------------------------------------------------
// Phase 3: y[o] = Wy[o,:] . h_last + by[o]
// ---------------------------------------------------------------------------
__global__ void __launch_bounds__(O_DIM)
out_gemv(const float* __restrict__ hs, const float* __restrict__ Wy,
         const float* __restrict__ by, float* __restrict__ y) {
  __shared__ float hl[H_DIM];
  for (int i = threadIdx.x; i < H_DIM; i += O_DIM)
    hl[i] = hs[(size_t)i * T_DIM + (T_DIM - 1)];   // h_last = hs[:, T-1]
  __syncthreads();
  const int o = threadIdx.x;
  float s = by[o];
  const float* wr = Wy + (size_t)o * H_DIM;
#pragma unroll 8
  for (int k = 0; k < H_DIM; ++k) s = fmaf(wr[k], hl[k], s);
  y[o] = s;
}

__global__ void init_ctrl(int* bar) { bar[0] = 0; bar[1] = 0; }

// ---------------------------------------------------------------------------
extern "C" void kernel_launch(void* const* d_in, const int* in_sizes, int n_in,
                              void* d_out, int out_size, void* d_ws, size_t ws_size,
                              hipStream_t stream) {
  const float* x  = (const float*)d_in[0];
  const float* Wx = (const float*)d_in[1];
  const float* Wh = (const float*)d_in[2];
  const float* Wy = (const float*)d_in[3];
  const float* bh = (const float*)d_in[4];
  const float* by = (const float*)d_in[5];

  float* hs_out = (float*)d_out;                           // [H*T]
  float* y_out  = hs_out + (size_t)H_DIM * T_DIM;          // [O]

  float* xw     = (float*)d_ws;                            // [T*H] scratch
  float* hshare = xw + (size_t)T_DIM * H_DIM;              // [2*H]
  int*   bar    = (int*)(hshare + 2 * H_DIM);              // [2]

  const size_t lds_bytes = (size_t)(ROWS_PER_BLK * H_DIM + H_DIM) * sizeof(float);
  (void)hipFuncSetAttribute((const void*)rnn_scan,
                            hipFuncAttributeMaxDynamicSharedMemorySize,
                            (int)lds_bytes);               // allow 260KB LDS

  init_ctrl<<<1, 1, 0, stream>>>(bar);
  xw_gemm<<<XW_GRID, 256, 0, stream>>>(x, Wx, bh, xw);
  rnn_scan<<<NBLK, SCAN_THREADS, lds_bytes, stream>>>(xw, Wh, hs_out, hshare, bar);
  out_gemv<<<1, O_DIM, 0, stream>>>(hs_out, Wy, by, y_out);
}